// MLPPredictor_89953795047574
// MI455X (gfx1250) — compile-verified
//
#include <hip/hip_runtime.h>

typedef float v2f __attribute__((ext_vector_type(2)));
typedef float v4f __attribute__((ext_vector_type(4)));
typedef float v8f __attribute__((ext_vector_type(8)));

#define F_DIM 128   // node feature dim
#define OUT_DIM 16  // output dim
#define PSTRIDE 32  // P row: [u-proj 0:16 | v-proj 16:32]

// ---------------------------------------------------------------------------
// Phase 1: per-node projection  P[n][0:16]  = h[n] . W[:, 0:128]^T
//                               P[n][16:32] = h[n] . W[:, 128:256]^T
// One wave computes a 16-row x 32-col tile using V_WMMA_F32_16X16X4_F32.
// A fragment (16x4 f32, 2 VGPRs): lane l -> A[l%16][kb + j + 2*(l>=16)]
// B fragment (4x16 f32, 2 VGPRs): lane l -> B[kb + j + 2*(l>=16)][l%16]
// C/D (16x16 f32, 8 VGPRs): vgpr r, lane l -> row r + 8*(l>=16), col l%16
// ---------------------------------------------------------------------------
__global__ __launch_bounds__(256) void node_project_wmma(
    const float* __restrict__ h,   // [N,128]
    const float* __restrict__ W,   // [16,256]
    float* __restrict__ P,         // [N,32]
    int N)
{
  const int lane = threadIdx.x & 31;
  const int wave = threadIdx.x >> 5;            // 0..7
  const int rowBase = blockIdx.x * 128 + wave * 16;
  if (rowBase >= N) return;                     // uniform per-wave exit

  const int m   = lane & 15;                    // A row / B col within tile
  const int hi  = lane >> 4;                    // upper half-wave?
  const int klo = hi * 2;                       // k sub-offset for this lane

  int rowA = rowBase + m;
  if (rowA >= N) rowA = N - 1;                  // clamp tail: EXEC stays all-1s
  const float* __restrict__ hrow = h + (size_t)rowA * F_DIM;
  const float* __restrict__ wrow = W + (size_t)m * (2 * F_DIM);  // W[n][.]

  v8f c0 = {};  // columns 0..15  (W_u)
  v8f c1 = {};  // columns 16..31 (W_v)

#pragma unroll 8
  for (int kb = 0; kb < F_DIM; kb += 4) {
    const int k = kb + klo;                     // even -> 8B aligned
    // h is streamed exactly once: non-temporal so it doesn't evict W/P.
    v2f a  = __builtin_nontemporal_load((const v2f*)(hrow + k));
    v2f b0 = *(const v2f*)(wrow + k);           // W: hot, keep cached
    v2f b1 = *(const v2f*)(wrow + F_DIM + k);
    // (neg_a, A, neg_b, B, c_mod, C, reuse_a, reuse_b)
    c0 = __builtin_amdgcn_wmma_f32_16x16x4_f32(false, a, false, b0,
                                               (short)0, c0, false, false);
    c1 = __builtin_amdgcn_wmma_f32_16x16x4_f32(false, a, false, b1,
                                               (short)0, c1, false, false);
  }

  // Store epilogue: branch-free fast path for full tiles (all but the last).
  float* __restrict__ prow0 = P + (size_t)(rowBase + hi * 8) * PSTRIDE + m;
  if (rowBase + 16 <= N) {
#pragma unroll
    for (int r = 0; r < 8; ++r) {
      prow0[r * PSTRIDE]           = c0[r];     // immediate offsets r*128B
      prow0[r * PSTRIDE + OUT_DIM] = c1[r];     // .. +64B
    }
  } else {
#pragma unroll
    for (int r = 0; r < 8; ++r) {
      if (rowBase + hi * 8 + r < N) {
        prow0[r * PSTRIDE]           = c0[r];
        prow0[r * PSTRIDE + OUT_DIM] = c1[r];
      }
    }
  }
}

// ---------------------------------------------------------------------------
// Phase 2: per-edge gather + add.  4 lanes per edge, v4f per lane.
// out[e][o] = P[src[e]][o] + P[dst[e]][16+o] + b[o]
// P (12.8 MB) stays L2-resident (out is written non-temporally); this phase
// is output-bandwidth bound (~41 MB stores + ~5 MB index reads).
// ---------------------------------------------------------------------------
__global__ __launch_bounds__(256) void edge_gather_add(
    const int* __restrict__ src,
    const int* __restrict__ dst,
    const float* __restrict__ P,   // [N,32]
    const float* __restrict__ b,   // [16]
    float* __restrict__ out,       // [E,16]
    int E)
{
  const int tid = blockIdx.x * blockDim.x + threadIdx.x;
  const int e = tid >> 2;
  if (e >= E) return;
  const int q = (tid & 3) * 4;

  const int s = src[e];
  const int d = dst[e];

  const v4f u  = *(const v4f*)(P + (size_t)s * PSTRIDE + q);
  const v4f v  = *(const v4f*)(P + (size_t)d * PSTRIDE + OUT_DIM + q);
  const v4f bv = *(const v4f*)(b + q);

  const v4f r = u + v + bv;

  // Write-once stream: non-temporal store keeps P resident in L2.
  __builtin_nontemporal_store(r, (v4f*)(out + (size_t)e * OUT_DIM + q));
}

extern "C" void kernel_launch(void* const* d_in, const int* in_sizes, int n_in,
                              void* d_out, int out_size, void* d_ws, size_t ws_size,
                              hipStream_t stream) {
  const float* h   = (const float*)d_in[0];   // [N,128] f32
  const int*   src = (const int*)d_in[1];     // [E] int
  const int*   dst = (const int*)d_in[2];     // [E] int
  const float* W   = (const float*)d_in[3];   // [16,256] f32
  const float* b   = (const float*)d_in[4];   // [16] f32
  float* out = (float*)d_out;                 // [E,16] f32
  float* P   = (float*)d_ws;                  // [N,32] f32 scratch (12.8 MB)

  const int N = in_sizes[0] / F_DIM;
  const int E = in_sizes[1];

  // Phase 1: node projection GEMM (WMMA f32 16x16x4)
  const int grid1 = (N + 127) / 128;          // 128 rows per block (8 waves)
  node_project_wmma<<<grid1, 256, 0, stream>>>(h, W, P, N);

  // Phase 2: edge gather + add (4 lanes / edge)
  const long long threads2 = (long long)E * 4;
  const int grid2 = (int)((threads2 + 255) / 256);
  edge_gather_add<<<grid2, 256, 0, stream>>>(src, dst, P, b, out, E);
}